// HierarchicalGrid_17514876634248
// MI455X (gfx1250) — compile-verified
//
#include <hip/hip_runtime.h>
#include <stdint.h>

#define FEAT 8
#define BLOCK 256

// ---- gfx1250 async store-from-LDS availability ------------------------------
#if defined(__has_builtin)
#if __has_builtin(__builtin_amdgcn_global_store_async_from_lds_b128) && \
    __has_builtin(__builtin_amdgcn_s_wait_asynccnt)
#define HG_ASYNC 1
#endif
#endif
#ifndef HG_ASYNC
#define HG_ASYNC 0
#endif

// Builtin parameter type per hipcc diagnostic:
//   __attribute__((__vector_size__(4*sizeof(int)))) int __device__ *
typedef int v4i __attribute__((vector_size(4 * sizeof(int))));
typedef __attribute__((address_space(1))) v4i g_v4i;  // global (target AS1)
typedef __attribute__((address_space(3))) v4i l_v4i;  // LDS    (target AS3)

// Integer detours avoid pointer addrspace-cast diagnostics. Flat LDS addresses
// carry the LDS byte offset in the low 32 bits (ISA 10.2).
static __device__ __forceinline__ g_v4i* as_global_v4i(const void* p) {
  return (g_v4i*)(uintptr_t)p;
}
static __device__ __forceinline__ l_v4i* as_lds_v4i(const void* p) {
  return (l_v4i*)(uint32_t)(uintptr_t)p;
}

struct Corner8 {
  int   i[8];           // corner offsets in units of FEAT floats
  float fx, fy, fz;
};

static __device__ __forceinline__ Corner8 make_corners(int R, float wx, float wy,
                                                       float wz) {
  const float Rm1 = (float)(R - 1);
  // reference: g = w*(R-1); pix = (g+1)*0.5*(R-1); clip to [0, R-1]
  float px = fminf(fmaxf((wx * Rm1 + 1.0f) * 0.5f * Rm1, 0.0f), Rm1);
  float py = fminf(fmaxf((wy * Rm1 + 1.0f) * 0.5f * Rm1, 0.0f), Rm1);
  float pz = fminf(fmaxf((wz * Rm1 + 1.0f) * 0.5f * Rm1, 0.0f), Rm1);
  int x0 = (int)px, y0 = (int)py, z0 = (int)pz;  // px,py,pz >= 0 -> trunc==floor
  Corner8 c;
  c.fx = px - (float)x0;
  c.fy = py - (float)y0;
  c.fz = pz - (float)z0;
  int x1 = (x0 + 1 < R) ? x0 + 1 : R - 1;
  int y1 = (y0 + 1 < R) ? y0 + 1 : R - 1;
  int z1 = (z0 + 1 < R) ? z0 + 1 : R - 1;
  int r00 = (z0 * R + y0) * R;
  int r01 = (z0 * R + y1) * R;
  int r10 = (z1 * R + y0) * R;
  int r11 = (z1 * R + y1) * R;
  c.i[0] = r00 + x0; c.i[1] = r00 + x1;
  c.i[2] = r01 + x0; c.i[3] = r01 + x1;
  c.i[4] = r10 + x0; c.i[5] = r10 + x1;
  c.i[6] = r11 + x0; c.i[7] = r11 + x1;
  return c;
}

static __device__ __forceinline__ float4 wmix(float4 a, float4 b, float wa,
                                              float wb) {
  return make_float4(a.x * wa + b.x * wb, a.y * wa + b.y * wb,
                     a.z * wa + b.z * wb, a.w * wa + b.w * wb);
}

static __device__ __forceinline__ void sample8(const float* __restrict__ g,
                                               const Corner8& c, float4& lo,
                                               float4& hi) {
  const float4* g4 = (const float4*)g;  // corner i -> float4s [2i, 2i+1]
  const float xa = 1.0f - c.fx, xb = c.fx;
  float4 c00l = wmix(g4[(size_t)c.i[0] * 2 + 0], g4[(size_t)c.i[1] * 2 + 0], xa, xb);
  float4 c00h = wmix(g4[(size_t)c.i[0] * 2 + 1], g4[(size_t)c.i[1] * 2 + 1], xa, xb);
  float4 c01l = wmix(g4[(size_t)c.i[2] * 2 + 0], g4[(size_t)c.i[3] * 2 + 0], xa, xb);
  float4 c01h = wmix(g4[(size_t)c.i[2] * 2 + 1], g4[(size_t)c.i[3] * 2 + 1], xa, xb);
  float4 c10l = wmix(g4[(size_t)c.i[4] * 2 + 0], g4[(size_t)c.i[5] * 2 + 0], xa, xb);
  float4 c10h = wmix(g4[(size_t)c.i[4] * 2 + 1], g4[(size_t)c.i[5] * 2 + 1], xa, xb);
  float4 c11l = wmix(g4[(size_t)c.i[6] * 2 + 0], g4[(size_t)c.i[7] * 2 + 0], xa, xb);
  float4 c11h = wmix(g4[(size_t)c.i[6] * 2 + 1], g4[(size_t)c.i[7] * 2 + 1], xa, xb);
  const float ya = 1.0f - c.fy, yb = c.fy;
  float4 c0l = wmix(c00l, c01l, ya, yb);
  float4 c0h = wmix(c00h, c01h, ya, yb);
  float4 c1l = wmix(c10l, c11l, ya, yb);
  float4 c1h = wmix(c10h, c11h, ya, yb);
  const float za = 1.0f - c.fz, zb = c.fz;
  lo = wmix(c0l, c1l, za, zb);
  hi = wmix(c0h, c1h, za, zb);
}

__global__ __launch_bounds__(BLOCK) void hg_sample_kernel(
    const float* __restrict__ coords, const float* __restrict__ g0,
    const float* __restrict__ g1, const float* __restrict__ g2,
    const float* __restrict__ g3, float* __restrict__ out, int N) {
  // Point-major staging tile, padded 8->9 float4 slots/point to break the
  // 128B-stride bank pathology (64 banks) on the ds_store_b128 writes.
  __shared__ float4 sbuf[BLOCK * 9];

  const int tid = threadIdx.x;
  const int p = blockIdx.x * BLOCK + tid;
  const bool valid = (p < N);

  float wx = 0.0f, wy = 0.0f, wz = 0.0f;
  if (valid) {
    wx = coords[(size_t)p * 3 + 0];
    wy = coords[(size_t)p * 3 + 1];
    wz = coords[(size_t)p * 3 + 2];
  }

  const float* gs[4] = {g0, g1, g2, g3};

  // Level 3 (256^3 = 512 MB) is the only HBM-resident grid: compute its corner
  // addresses up front and prefetch them so the random-access HBM latency is
  // hidden behind the L2-resident levels 0-2.
  Corner8 c3 = make_corners(256, wx, wy, wz);
  if (valid) {
#pragma unroll
    for (int k = 0; k < 8; ++k)
      __builtin_prefetch(g3 + (size_t)c3.i[k] * FEAT, 0, 0);
  }

#pragma unroll
  for (int lvl = 0; lvl < 4; ++lvl) {
    const int R = 32 << lvl;
    float4 lo = make_float4(0.f, 0.f, 0.f, 0.f), hi = lo;
    if (valid) {
      Corner8 c = (lvl == 3) ? c3 : make_corners(R, wx, wy, wz);
      sample8(gs[lvl], c, lo, hi);
    }
    sbuf[tid * 9 + lvl * 2 + 0] = lo;
    sbuf[tid * 9 + lvl * 2 + 1] = hi;
  }

  __syncthreads();

  // Coalesced write-out: each pass stores 256 consecutive float4s (4 KB of
  // full-line B128 stores) straight from LDS via the async store path.
  const size_t blockBase4 = (size_t)blockIdx.x * (BLOCK * 8);  // float4 units
  const size_t total4 = (size_t)N * 8;
#pragma unroll
  for (int k = 0; k < 8; ++k) {
    const int g4i = k * BLOCK + tid;
    const int pp = g4i >> 3;  // local point
    const int j = g4i & 7;    // float4 slot within point
    const size_t o4 = blockBase4 + (size_t)g4i;
    if (o4 < total4) {
#if HG_ASYNC
      __builtin_amdgcn_global_store_async_from_lds_b128(
          as_global_v4i((const float4*)out + o4), as_lds_v4i(&sbuf[pp * 9 + j]),
          0, 0);
#else
      ((float4*)out)[o4] = sbuf[pp * 9 + j];
#endif
    }
  }
#if HG_ASYNC
  __builtin_amdgcn_s_wait_asynccnt(0);
#endif
}

extern "C" void kernel_launch(void* const* d_in, const int* in_sizes, int n_in,
                              void* d_out, int out_size, void* d_ws,
                              size_t ws_size, hipStream_t stream) {
  const float* coords = (const float*)d_in[0];
  const float* g0 = (const float*)d_in[1];
  const float* g1 = (const float*)d_in[2];
  const float* g2 = (const float*)d_in[3];
  const float* g3 = (const float*)d_in[4];
  float* out = (float*)d_out;

  const int N = in_sizes[0] / 3;  // 2,097,152 points
  const int blocks = (N + BLOCK - 1) / BLOCK;
  hipLaunchKernelGGL(hg_sample_kernel, dim3(blocks), dim3(BLOCK), 0, stream,
                     coords, g0, g1, g2, g3, out, N);
}